// NormalizedGCNConv_80607946211867
// MI455X (gfx1250) — compile-verified
//
#include <hip/hip_runtime.h>

typedef __attribute__((ext_vector_type(2))) float v2f;
typedef __attribute__((ext_vector_type(8))) float v8f;

#define DFEAT 128  // d_in == d_out == 128 in this problem

// ---------------- zero / init ----------------
__global__ void gcn_zero_out(float* __restrict__ out, int total) {
    int i = blockIdx.x * blockDim.x + threadIdx.x;
    if (i < total) out[i] = 0.0f;
}

__global__ void gcn_init_deg(float* __restrict__ deg, int n) {
    int i = blockIdx.x * blockDim.x + threadIdx.x;
    if (i < n) deg[i] = 1.0f;  // self-loop contributes 1 to every node's degree
}

// ---------------- degree count ----------------
__global__ void gcn_deg(const long long* __restrict__ ei, float* __restrict__ deg, int E) {
    int e = blockIdx.x * blockDim.x + threadIdx.x;
    if (e < E) {
        int d = (int)ei[(size_t)E + e];  // dst row of edge_index
        unsafeAtomicAdd(&deg[d], 1.0f);
    }
}

__global__ void gcn_rsqrt(float* __restrict__ deg, int n) {
    int i = blockIdx.x * blockDim.x + threadIdx.x;
    if (i < n) deg[i] = rsqrtf(deg[i]);  // deg >= 1 always (self loops)
}

// ---------------- h = x @ W^T via V_WMMA_F32_16X16X4_F32 ----------------
// One wave computes a full 16x128 row block of h: 8 accumulator tiles share
// each A fragment, so x is read exactly once from HBM and W (64 KB,
// L2-resident) supplies the B fragments.
// A (16x4 f32): lanes 0-15 hold rows M=0..15 with K={k,k+1}; lanes 16-31 K={k+2,k+3}.
// B (4x16 f32): B[k][nn] = W[n0+nn][k]; symmetric lane layout.
// C/D (16x16 f32, 8 VGPRs): VGPR r -> M = r (lanes 0-15) / r+8 (lanes 16-31), N = lane%16.
__global__ void __launch_bounds__(128) gcn_gemm_wmma(const float* __restrict__ x,
                                                     const float* __restrict__ W,
                                                     float* __restrict__ h, int n) {
    const int lane = threadIdx.x & 31;
    const int wave = threadIdx.x >> 5;
    const int mt   = blockIdx.x * 4 + wave;      // one M-tile per wave
    const int m0   = mt * 16;
    if (m0 >= n) return;                         // uniform per wave

    const int half = lane >> 4;                  // 0 or 1
    const int lr   = lane & 15;

    int rowA = m0 + lr;
    if (rowA >= n) rowA = n - 1;                 // clamp (n % 16 == 0 here anyway)
    const float* __restrict__ xrow  = x + (size_t)rowA * DFEAT;
    const float* __restrict__ wbase = W + (size_t)lr * DFEAT;  // + t*16*DFEAT per N-tile

    v8f c[8];
#pragma unroll
    for (int t = 0; t < 8; ++t) c[t] = (v8f){};

    for (int k = 0; k < DFEAT; k += 4) {
        const int ko = k + 2 * half;
        const v2f a = *(const v2f*)(xrow + ko);
#pragma unroll
        for (int t = 0; t < 8; ++t) {
            const v2f b = *(const v2f*)(wbase + (size_t)t * 16 * DFEAT + ko);
            c[t] = __builtin_amdgcn_wmma_f32_16x16x4_f32(
                       /*neg_a=*/false, a, /*neg_b=*/false, b,
                       /*c_mod=*/(short)0, c[t], /*reuse_a=*/false, /*reuse_b=*/false);
        }
    }

    // store: row m0+r+8*half, cols t*16+lr for t=0..7
#pragma unroll
    for (int r = 0; r < 8; ++r) {
        const int row = m0 + r + 8 * half;
        if (row < n) {
            float* __restrict__ hrow = h + (size_t)row * DFEAT + lr;
#pragma unroll
            for (int t = 0; t < 8; ++t) hrow[t * 16] = c[t][r];
        }
    }
}

// ---------------- edge scatter: out[dst] += h[src] * norm ----------------
// One wave per edge; lane l owns columns [4l, 4l+4): float4 gather + 4 f32 atomics.
__global__ void __launch_bounds__(256) gcn_scatter(const long long* __restrict__ ei,
                                                   const float* __restrict__ h,
                                                   const float* __restrict__ dinv,
                                                   float* __restrict__ out, int E) {
    const long long t = (long long)blockIdx.x * blockDim.x + threadIdx.x;
    const int e    = (int)(t >> 5);
    const int lane = (int)(t & 31);
    if (e >= E) return;
    const int s = (int)ei[e];
    const int d = (int)ei[(size_t)E + e];
    const float norm = dinv[s] * dinv[d];
    const float4 hv = *(const float4*)(h + (size_t)s * DFEAT + lane * 4);
    float* o = out + (size_t)d * DFEAT + lane * 4;
    unsafeAtomicAdd(o + 0, hv.x * norm);
    unsafeAtomicAdd(o + 1, hv.y * norm);
    unsafeAtomicAdd(o + 2, hv.z * norm);
    unsafeAtomicAdd(o + 3, hv.w * norm);
}

// ---------------- self-loop + bias ----------------
__global__ void gcn_finish(const float* __restrict__ h, const float* __restrict__ dinv,
                           const float* __restrict__ bias, float* __restrict__ out, int n) {
    const int t = blockIdx.x * blockDim.x + threadIdx.x;  // over n * DFEAT/4
    const int i  = t / (DFEAT / 4);
    const int j4 = (t % (DFEAT / 4)) * 4;
    if (i >= n) return;
    const float w = dinv[i] * dinv[i];
    const float4 hv = *(const float4*)(h + (size_t)i * DFEAT + j4);
    float* o = out + (size_t)i * DFEAT + j4;
    o[0] += hv.x * w + bias[j4 + 0];
    o[1] += hv.y * w + bias[j4 + 1];
    o[2] += hv.z * w + bias[j4 + 2];
    o[3] += hv.w * w + bias[j4 + 3];
}

extern "C" void kernel_launch(void* const* d_in, const int* in_sizes, int n_in,
                              void* d_out, int out_size, void* d_ws, size_t ws_size,
                              hipStream_t stream) {
    const float*     x    = (const float*)d_in[0];
    const long long* ei   = (const long long*)d_in[1];
    const float*     W    = (const float*)d_in[2];
    const float*     bias = (const float*)d_in[3];
    float*           out  = (float*)d_out;

    const int dout = in_sizes[3];
    const int din  = in_sizes[2] / dout;
    const int n    = in_sizes[0] / din;
    const int E    = in_sizes[1] / 2;

    // workspace: h [n*dout] f32, then deg/dinv [n] f32
    float* h   = (float*)d_ws;
    float* deg = h + (size_t)n * dout;

    const int total = n * dout;
    gcn_zero_out<<<(total + 255) / 256, 256, 0, stream>>>(out, total);
    gcn_init_deg<<<(n + 255) / 256, 256, 0, stream>>>(deg, n);
    gcn_deg<<<(E + 255) / 256, 256, 0, stream>>>(ei, deg, E);
    gcn_rsqrt<<<(n + 255) / 256, 256, 0, stream>>>(deg, n);

    const int mtiles = (n + 15) / 16;            // one wave per 16-row block
    gcn_gemm_wmma<<<(mtiles + 3) / 4, 128, 0, stream>>>(x, W, h, n);

    const long long sthreads = (long long)E * 32;
    gcn_scatter<<<(int)((sthreads + 255) / 256), 256, 0, stream>>>(ei, h, deg, out, E);

    const int fthreads = n * (dout / 4);
    gcn_finish<<<(fthreads + 255) / 256, 256, 0, stream>>>(h, deg, bias, out, n);
}